// GraphSAGE_1967095021810
// MI455X (gfx1250) — compile-verified
//
#include <hip/hip_runtime.h>
#include <math.h>

#define N_NODES   100000
#define N_EDGES   1600000
#define F_IN      50
#define N_CLASSES 16

typedef __attribute__((ext_vector_type(2))) float v2f;
typedef __attribute__((ext_vector_type(8))) float v8f;

// ---------------------------------------------------------------------------
// Kernel 0: zero the workspace (agg_sum [N,50] + deg [N]) every launch —
// harness does not re-zero scratch between replays and our atomics accumulate.
// ---------------------------------------------------------------------------
__global__ void sage_zero_ws(float* __restrict__ ws, int n) {
    int i = blockIdx.x * blockDim.x + threadIdx.x;
    int stride = gridDim.x * blockDim.x;
    for (; i < n; i += stride) ws[i] = 0.0f;
}

// ---------------------------------------------------------------------------
// Kernel 1: edge-parallel segment-sum. 4 edges per 256-thread block; lane
// f = t&63 handles feature f (f<50); lane f==50 bumps the degree counter.
// Consecutive lanes touch consecutive floats of one row -> coalesced L2
// gathers and coalesced non-returning f32 atomics (x and agg_sum are both
// L2-resident at 20 MB each on a 192 MB L2).
// ---------------------------------------------------------------------------
__global__ __launch_bounds__(256)
void sage_edge_agg(const int* __restrict__ ei, const float* __restrict__ x,
                   float* __restrict__ aggsum, float* __restrict__ deg) {
    int e = blockIdx.x * 4 + (threadIdx.x >> 6);
    int f = threadIdx.x & 63;
    if (e >= N_EDGES) return;
    int src = ei[e];             // edge_index[0][e]
    int tgt = ei[N_EDGES + e];   // edge_index[1][e]
    if (f < F_IN) {
        unsafeAtomicAdd(&aggsum[tgt * F_IN + f], x[src * F_IN + f]);
    } else if (f == F_IN) {
        unsafeAtomicAdd(&deg[tgt], 1.0f);
    }
}

// ---------------------------------------------------------------------------
// Kernel 2: fused SAGE linear + bias + log_softmax using V_WMMA_F32_16X16X4_F32.
// One wave = one 16-node x 16-class tile. K (=50, padded to 52) tiled by 4;
// both GEMMs (agg@Wl^T and x@Wr^T) accumulate into one 8-VGPR f32 C/D tile.
// ---------------------------------------------------------------------------
__global__ __launch_bounds__(256)
void sage_gemm_softmax(const float* __restrict__ aggsum, const float* __restrict__ deg,
                       const float* __restrict__ x,
                       const float* __restrict__ Wl, const float* __restrict__ bl,
                       const float* __restrict__ Wr,
                       float* __restrict__ out_logp, float* __restrict__ out_lin) {
    const int lane = threadIdx.x & 31;
    const int wave = threadIdx.x >> 5;
    const int tile = blockIdx.x * 8 + wave;   // 16 nodes per tile
    const int base = tile * 16;
    if (base >= N_NODES) return;              // wave-uniform exit (EXEC intact)

    const int m  = lane & 15;                 // A-matrix row within tile
    const int hi = lane >> 4;                 // half-wave selector
    const int n  = lane & 15;                 // class column for B/C/D

    int nodeA = base + m;
    if (nodeA >= N_NODES) nodeA = N_NODES - 1;   // never triggers (N%16==0), safe anyway
    float r = deg[nodeA];
    r = 1.0f / fmaxf(r, 1.0f);                   // mean normalization

    v8f acc = {};
    #pragma unroll
    for (int k0 = 0; k0 < F_IN; k0 += 4) {
        const int kx = k0 + 2 * hi;      // .x element's K index (layout: V0 = K0,K2 halves)
        const int ky = kx + 1;           // .y element's K index (V1 = K1,K3 halves)
        const bool vx = (kx < F_IN);
        const bool vy = (ky < F_IN);

        v2f a_agg, a_x, b_l, b_r;
        a_agg.x = vx ? aggsum[nodeA * F_IN + kx] * r : 0.0f;
        a_agg.y = vy ? aggsum[nodeA * F_IN + ky] * r : 0.0f;
        a_x.x   = vx ? x[nodeA * F_IN + kx] : 0.0f;
        a_x.y   = vy ? x[nodeA * F_IN + ky] : 0.0f;
        // B[k][n] = W[n][k]  (W stored row-major [C, F])
        b_l.x   = vx ? Wl[n * F_IN + kx] : 0.0f;
        b_l.y   = vy ? Wl[n * F_IN + ky] : 0.0f;
        b_r.x   = vx ? Wr[n * F_IN + kx] : 0.0f;
        b_r.y   = vy ? Wr[n * F_IN + ky] : 0.0f;

        acc = __builtin_amdgcn_wmma_f32_16x16x4_f32(
                  false, a_agg, false, b_l, (short)0, acc, false, false);
        acc = __builtin_amdgcn_wmma_f32_16x16x4_f32(
                  false, a_x,   false, b_r, (short)0, acc, false, false);
    }

    const float bias = bl[n];

    // C/D layout: VGPR reg -> row m = reg + 8*hi; lane -> class n.
    // log-softmax across the 16 classes = 16-lane reduction per half-wave.
    #pragma unroll
    for (int reg = 0; reg < 8; ++reg) {
        float v = acc[reg] + bias;
        int node = base + reg + 8 * hi;

        float mx = v;
        #pragma unroll
        for (int off = 8; off >= 1; off >>= 1)
            mx = fmaxf(mx, __shfl_xor(mx, off, 16));
        float ex = __expf(v - mx);
        float s = ex;
        #pragma unroll
        for (int off = 8; off >= 1; off >>= 1)
            s += __shfl_xor(s, off, 16);
        float lp = v - mx - __logf(s);

        if (node < N_NODES) {
            out_lin [node * N_CLASSES + n] = v;
            out_logp[node * N_CLASSES + n] = lp;
        }
    }
}

// ---------------------------------------------------------------------------
extern "C" void kernel_launch(void* const* d_in, const int* in_sizes, int n_in,
                              void* d_out, int out_size, void* d_ws, size_t ws_size,
                              hipStream_t stream) {
    const float* x  = (const float*)d_in[0];   // [N, 50]
    const int*   ei = (const int*)  d_in[1];   // [2, E]
    const float* Wl = (const float*)d_in[2];   // [16, 50]
    const float* bl = (const float*)d_in[3];   // [16]
    const float* Wr = (const float*)d_in[4];   // [16, 50]

    float* out  = (float*)d_out;
    float* logp = out;                                   // first output: logp [N,16]
    float* lin  = out + (size_t)N_NODES * N_CLASSES;     // second output: out  [N,16]

    float* aggsum = (float*)d_ws;                        // [N, 50]
    float* deg    = aggsum + (size_t)N_NODES * F_IN;     // [N]

    const int nzero = N_NODES * F_IN + N_NODES;
    sage_zero_ws<<<1024, 256, 0, stream>>>(aggsum, nzero);

    sage_edge_agg<<<(N_EDGES + 3) / 4, 256, 0, stream>>>(ei, x, aggsum, deg);

    const int ntiles  = (N_NODES + 15) / 16;   // 6250
    const int nblocks = (ntiles + 7) / 8;      // 782 (8 waves/block)
    sage_gemm_softmax<<<nblocks, 256, 0, stream>>>(aggsum, deg, x, Wl, bl, Wr, logp, lin);
}